// GATEncoder_36395552866783
// MI455X (gfx1250) — compile-verified
//
#include <hip/hip_runtime.h>
#include <math.h>

#define N_NODES 50000
#define N_EDGES 800000
#define F_IN    64
#define HEADS   8
#define HID     16
#define CTOT    128         // HEADS*HID == OUT == 128
#define NEG_SLOPE 0.2f
#define EPS_BN  1e-5f
#define BN_ROWS 250

typedef float v2f __attribute__((ext_vector_type(2)));
typedef float v8f __attribute__((ext_vector_type(8)));

// ---------------------------------------------------------------- utilities

__global__ void fill_kernel(float* __restrict__ p, float v, int n) {
    int i = blockIdx.x * blockDim.x + threadIdx.x;
    if (i < n) p[i] = v;
}

__device__ inline void atomicMaxF(float* addr, float val) {
    unsigned int* ua = reinterpret_cast<unsigned int*>(addr);
    unsigned int cur = __float_as_uint(*addr);
    while (__uint_as_float(cur) < val) {
        unsigned int prev = atomicCAS(ua, cur, __float_as_uint(val));
        if (prev == cur) break;
        cur = prev;
    }
}

// ---------------------------------------------------------------- WMMA GEMM
// C[M x NC] = A[M x K] @ B[K x NC] (+ bias[NC]); fp32 via V_WMMA_F32_16X16X4_F32.
// One wave per 16x16 output tile. M % 16 == 0, NC % 16 == 0, K % 4 == 0.
__global__ void gemm_wmma_f32(const float* __restrict__ A,
                              const float* __restrict__ B,
                              const float* __restrict__ bias,
                              float* __restrict__ C,
                              int M, int K, int NC) {
    int wave = (blockIdx.x * blockDim.x + threadIdx.x) >> 5;
    int lane = threadIdx.x & 31;
    int tilesN = NC >> 4;
    int tm = wave / tilesN;
    int tn = wave - tm * tilesN;
    if (tm * 16 >= M) return;                  // wave-uniform guard (EXEC stays full)

    int row   = tm * 16 + (lane & 15);         // A-fragment row (M)
    int col   = tn * 16 + (lane & 15);         // B-fragment col (N)
    int kgrp  = (lane >> 4) * 2;               // lanes 0-15 -> K+0/K+1, lanes 16-31 -> K+2/K+3

    v8f acc = {};
    for (int k0 = 0; k0 < K; k0 += 4) {
        int kk = k0 + kgrp;
        v2f a, b;
        a.x = A[row * K + kk];
        a.y = A[row * K + kk + 1];
        b.x = B[kk * NC + col];
        b.y = B[(kk + 1) * NC + col];
        acc = __builtin_amdgcn_wmma_f32_16x16x4_f32(
                  /*neg_a=*/false, a, /*neg_b=*/false, b,
                  /*c_mod=*/(short)0, acc, /*reuse_a=*/false, /*reuse_b=*/false);
    }

    float bv = bias ? bias[col] : 0.0f;
    int r0 = tm * 16 + (lane >> 4) * 8;        // C/D layout: VGPR i -> rows i and i+8
#pragma unroll
    for (int i = 0; i < 8; ++i)
        C[(r0 + i) * NC + col] = acc[i] + bv;
}

// ------------------------------------------------------- per-node attention dots
// asrc[n*H+h] = sum_c feat[n, h*Cc + c] * att_src[h*Cc + c]   (same for adst)
__global__ void attn_node(const float* __restrict__ feat,
                          const float* __restrict__ att_src,
                          const float* __restrict__ att_dst,
                          float* __restrict__ asrc, float* __restrict__ adst,
                          int H, int Cc) {
    int i = blockIdx.x * blockDim.x + threadIdx.x;
    if (i >= N_NODES * H) return;
    int n = i / H, h = i - n * H;
    const float* fp = feat + (size_t)n * CTOT + h * Cc;
    const float* as = att_src + h * Cc;
    const float* ad = att_dst + h * Cc;
    float s = 0.f, d = 0.f;
    for (int c = 0; c < Cc; ++c) { s += fp[c] * as[c]; d += fp[c] * ad[c]; }
    asrc[i] = s;
    adst[i] = d;
}

// ------------------------------------------------------- edge softmax passes
__global__ void edge_max(const int* __restrict__ src, const int* __restrict__ dst,
                         const float* __restrict__ asrc, const float* __restrict__ adst,
                         float* __restrict__ ebuf, float* __restrict__ m, int H) {
    int i = blockIdx.x * blockDim.x + threadIdx.x;
    if (i >= N_EDGES * H) return;
    int e = i / H, h = i - e * H;
    int s = src[e], d = dst[e];
    float v = asrc[s * H + h] + adst[d * H + h];
    v = (v >= 0.f) ? v : v * NEG_SLOPE;        // leaky_relu
    ebuf[i] = v;
    atomicMaxF(&m[d * H + h], v);
}

__global__ void edge_exp(const int* __restrict__ dst,
                         float* __restrict__ ebuf, const float* __restrict__ m,
                         float* __restrict__ den, int H) {
    int i = blockIdx.x * blockDim.x + threadIdx.x;
    if (i >= N_EDGES * H) return;
    int e = i / H, h = i - e * H;
    int d = dst[e];
    float mv = m[d * H + h];
    if (mv < -1e37f) mv = 0.0f;                // isfinite fixup (m init = -inf)
    float ex = expf(ebuf[i] - mv);
    ebuf[i] = ex;
    atomicAdd(&den[d * H + h], ex);
}

// one thread per (edge, feature); CH = channels per head, H*CH == 128
__global__ void edge_scatter(const int* __restrict__ src, const int* __restrict__ dst,
                             const float* __restrict__ ebuf, const float* __restrict__ den,
                             const float* __restrict__ feat, float* __restrict__ out,
                             int H, int CH) {
    int i = blockIdx.x * blockDim.x + threadIdx.x;
    if (i >= N_EDGES * CTOT) return;
    int e = i >> 7;
    int f = i & 127;
    int h = f / CH;
    int s = src[e], d = dst[e];
    float alpha = ebuf[e * H + h] / (den[d * H + h] + 1e-16f);
    atomicAdd(&out[(size_t)d * CTOT + f], alpha * feat[(size_t)s * CTOT + f]);
}

// ------------------------------------------------------- fused residual + BN
// v = gat[n,f] + bias[f] + resid[n,f]; store v in place; accumulate sum / sumsq.
__global__ void bn_stats(float* __restrict__ pre,
                         const float* __restrict__ bias,
                         const float* __restrict__ resid,
                         float* __restrict__ sum, float* __restrict__ sumsq) {
    int f  = threadIdx.x;                      // 0..127
    int r0 = blockIdx.x * BN_ROWS;
    float bv = bias[f];
    float s = 0.f, s2 = 0.f;
    for (int r = 0; r < BN_ROWS; ++r) {
        int n = r0 + r;
        if (n >= N_NODES) break;
        size_t idx = (size_t)n * CTOT + f;
        float v = pre[idx] + bv + resid[idx];
        pre[idx] = v;
        s += v; s2 += v * v;
    }
    atomicAdd(&sum[f], s);
    atomicAdd(&sumsq[f], s2);
}

__global__ void bn_final(const float* __restrict__ sum, const float* __restrict__ sumsq,
                         float* __restrict__ mu, float* __restrict__ inv) {
    int f = threadIdx.x;
    float m = sum[f] * (1.0f / N_NODES);
    float var = sumsq[f] * (1.0f / N_NODES) - m * m;   // biased variance
    mu[f]  = m;
    inv[f] = rsqrtf(var + EPS_BN);
}

__global__ void bn_apply(const float* __restrict__ pre,
                         const float* __restrict__ mu, const float* __restrict__ inv,
                         const float* __restrict__ g, const float* __restrict__ b,
                         float* __restrict__ out) {
    int i = blockIdx.x * blockDim.x + threadIdx.x;
    if (i >= N_NODES * CTOT) return;
    int f = i & 127;
    out[i] = (pre[i] - mu[f]) * inv[f] * g[f] + b[f];
}

// ---------------------------------------------------------------- launcher

static inline int cdiv(long a, long b) { return (int)((a + b - 1) / b); }

extern "C" void kernel_launch(void* const* d_in, const int* in_sizes, int n_in,
                              void* d_out, int out_size, void* d_ws, size_t ws_size,
                              hipStream_t stream) {
    const float* x        = (const float*)d_in[0];
    const int*   ei       = (const int*)  d_in[1];
    const float* W1       = (const float*)d_in[2];
    const float* att_src1 = (const float*)d_in[3];
    const float* att_dst1 = (const float*)d_in[4];
    const float* b1       = (const float*)d_in[5];
    const float* res1_W   = (const float*)d_in[6];
    const float* res1_b   = (const float*)d_in[7];
    const float* bn1_g    = (const float*)d_in[8];
    const float* bn1_b    = (const float*)d_in[9];
    const float* W3       = (const float*)d_in[10];
    const float* att_src3 = (const float*)d_in[11];
    const float* att_dst3 = (const float*)d_in[12];
    const float* b3       = (const float*)d_in[13];
    const float* res3_W   = (const float*)d_in[14];
    const float* res3_b   = (const float*)d_in[15];
    const float* bn3_g    = (const float*)d_in[16];
    const float* bn3_b    = (const float*)d_in[17];

    const int* src = ei;
    const int* dst = ei + N_EDGES;
    float* out = (float*)d_out;

    // workspace layout (floats)
    float* ws = (float*)d_ws;
    size_t o = 0;
    float* hfeat = ws + o; o += (size_t)N_NODES * CTOT;   // layer1 h ; reused as h2
    float* xres  = ws + o; o += (size_t)N_NODES * CTOT;   // x @ res1_W + res1_b
    float* gat1  = ws + o; o += (size_t)N_NODES * CTOT;   // gat accum / pre-BN ; reused as gat2
    float* h1bn  = ws + o; o += (size_t)N_NODES * CTOT;   // BN1 output
    float* ebuf1 = ws + o; o += (size_t)N_EDGES * HEADS;  // == N*128 floats; reused as res3out
    float* asrc1 = ws + o; o += (size_t)N_NODES * HEADS;
    float* adst1 = ws + o; o += (size_t)N_NODES * HEADS;
    float* m1    = ws + o; o += (size_t)N_NODES * HEADS;
    float* den1  = ws + o; o += (size_t)N_NODES * HEADS;
    float* ebuf2 = ws + o; o += (size_t)N_EDGES;
    float* sum   = ws + o; o += 128;
    float* sumsq = ws + o; o += 128;
    float* mu    = ws + o; o += 128;
    float* inv   = ws + o; o += 128;
    (void)ws_size; (void)in_sizes; (void)n_in; (void)out_size;

    const int GEMM_BLOCKS = (N_NODES / 16) * (CTOT / 16) / 8;   // 3125 blocks, 8 waves each
    const float NEG_INF = -__builtin_inff();

    // ---------------- layer 1 ----------------
    fill_kernel<<<cdiv((long)N_NODES * CTOT, 256), 256, 0, stream>>>(gat1, 0.f, N_NODES * CTOT);
    fill_kernel<<<cdiv((long)N_NODES * HEADS, 256), 256, 0, stream>>>(m1, NEG_INF, N_NODES * HEADS);
    fill_kernel<<<cdiv((long)N_NODES * HEADS, 256), 256, 0, stream>>>(den1, 0.f, N_NODES * HEADS);
    fill_kernel<<<1, 256, 0, stream>>>(sum, 0.f, 256);   // sum + sumsq contiguous

    gemm_wmma_f32<<<GEMM_BLOCKS, 256, 0, stream>>>(x, W1, nullptr, hfeat, N_NODES, F_IN, CTOT);
    gemm_wmma_f32<<<GEMM_BLOCKS, 256, 0, stream>>>(x, res1_W, res1_b, xres, N_NODES, F_IN, CTOT);

    attn_node<<<cdiv((long)N_NODES * HEADS, 256), 256, 0, stream>>>(
        hfeat, att_src1, att_dst1, asrc1, adst1, HEADS, HID);
    edge_max<<<cdiv((long)N_EDGES * HEADS, 256), 256, 0, stream>>>(
        src, dst, asrc1, adst1, ebuf1, m1, HEADS);
    edge_exp<<<cdiv((long)N_EDGES * HEADS, 256), 256, 0, stream>>>(
        dst, ebuf1, m1, den1, HEADS);
    edge_scatter<<<cdiv((long)N_EDGES * CTOT, 256), 256, 0, stream>>>(
        src, dst, ebuf1, den1, hfeat, gat1, HEADS, HID);

    bn_stats<<<cdiv(N_NODES, BN_ROWS), 128, 0, stream>>>(gat1, b1, xres, sum, sumsq);
    bn_final<<<1, 128, 0, stream>>>(sum, sumsq, mu, inv);
    bn_apply<<<cdiv((long)N_NODES * CTOT, 256), 256, 0, stream>>>(gat1, mu, inv, bn1_g, bn1_b, h1bn);

    // ---------------- layer 2 (reuse dead regions) ----------------
    float* h2      = hfeat;   // hfeat dead after scatter1
    float* gat2    = gat1;    // pre-BN1 dead after bn_apply
    float* res3out = ebuf1;   // ebuf1 dead after scatter1; E*8 == N*128 floats
    float* asrc2 = asrc1; float* adst2 = adst1; float* m2 = m1; float* den2 = den1;

    fill_kernel<<<cdiv((long)N_NODES * CTOT, 256), 256, 0, stream>>>(gat2, 0.f, N_NODES * CTOT);
    fill_kernel<<<cdiv((long)N_NODES, 256), 256, 0, stream>>>(m2, NEG_INF, N_NODES);
    fill_kernel<<<cdiv((long)N_NODES, 256), 256, 0, stream>>>(den2, 0.f, N_NODES);
    fill_kernel<<<1, 256, 0, stream>>>(sum, 0.f, 256);

    gemm_wmma_f32<<<GEMM_BLOCKS, 256, 0, stream>>>(h1bn, W3, nullptr, h2, N_NODES, CTOT, CTOT);
    gemm_wmma_f32<<<GEMM_BLOCKS, 256, 0, stream>>>(xres, res3_W, res3_b, res3out, N_NODES, CTOT, CTOT);

    attn_node<<<cdiv(N_NODES, 256), 256, 0, stream>>>(
        h2, att_src3, att_dst3, asrc2, adst2, 1, CTOT);
    edge_max<<<cdiv(N_EDGES, 256), 256, 0, stream>>>(src, dst, asrc2, adst2, ebuf2, m2, 1);
    edge_exp<<<cdiv(N_EDGES, 256), 256, 0, stream>>>(dst, ebuf2, m2, den2, 1);
    edge_scatter<<<cdiv((long)N_EDGES * CTOT, 256), 256, 0, stream>>>(
        src, dst, ebuf2, den2, h2, gat2, 1, CTOT);

    bn_stats<<<cdiv(N_NODES, BN_ROWS), 128, 0, stream>>>(gat2, b3, res3out, sum, sumsq);
    bn_final<<<1, 128, 0, stream>>>(sum, sumsq, mu, inv);
    bn_apply<<<cdiv((long)N_NODES * CTOT, 256), 256, 0, stream>>>(gat2, mu, inv, bn3_g, bn3_b, out);
}